// DropDim_71949292143167
// MI455X (gfx1250) — compile-verified
//
#include <hip/hip_runtime.h>

// DropDim: out[b,t,d] = h[b,t,d] * mask[b,d]   (B=16, T=4096, D=1024, fp32)
// Pure streaming kernel: ~537 MB traffic, 0.125 FLOP/byte -> HBM-bound (~23 us @ 23.3 TB/s).
// CDNA5 usage: async global->LDS staging of the broadcast mask row, NT cache hints
// for the h/out streams (537 MB vs 192 MB L2), b128 vector memops, prefetch.

typedef float v4f __attribute__((ext_vector_type(4)));
typedef int   v4i __attribute__((ext_vector_type(4)));

constexpr int Bsz = 16;
constexpr int Tsz = 4096;
constexpr int Dsz = 1024;

constexpr int THREADS           = 256;              // 8 wave32s
constexpr int BLOCKS_PER_SAMPLE = 128;
constexpr int ROWS_PER_BLOCK    = Tsz / BLOCKS_PER_SAMPLE;  // 32 rows of T per block
constexpr int PREFETCH_ROWS     = 8;

__global__ __launch_bounds__(THREADS)
void DropDim_kernel(const float* __restrict__ h,
                    const int*   __restrict__ mask,
                    float*       __restrict__ out) {
    __shared__ int smask[Dsz];  // 4 KB: one sample's mask row

    const int tid = threadIdx.x;
    const int b   = blockIdx.x / BLOCKS_PER_SAMPLE;  // sample index
    const int rb  = blockIdx.x % BLOCKS_PER_SAMPLE;  // row-block within sample

    // ---- Stage mask[b, :] (1024 x i32 = 4 KB) into LDS via CDNA5 async DMA ----
    // 256 lanes x 16 B each: one GLOBAL_LOAD_ASYNC_TO_LDS_B128 per lane (ASYNCcnt).
    {
        unsigned   lds_off = (unsigned)(size_t)&smask[tid * 4];       // low 32b of flat LDS addr == LDS byte offset
        const int* gsrc    = mask + (size_t)b * Dsz + (size_t)tid * 4;
        asm volatile("global_load_async_to_lds_b128 %0, %1, off"
                     :
                     : "v"(lds_off), "v"(gsrc)
                     : "memory");
        asm volatile("s_wait_asynccnt 0" ::: "memory");
    }
    __syncthreads();

    // Per-thread D-slice of the mask is loop-invariant over T: one ds_load_b128.
    const v4i mi = *(const v4i*)&smask[tid * 4];
    v4f m;
    m.x = mi.x ? 1.0f : 0.0f;
    m.y = mi.y ? 1.0f : 0.0f;
    m.z = mi.z ? 1.0f : 0.0f;
    m.w = mi.w ? 1.0f : 0.0f;

    // ---- Stream ROWS_PER_BLOCK rows of 4 KB: b128 NT load -> 4 muls -> b128 NT store ----
    const size_t base = ((size_t)b * Tsz + (size_t)rb * ROWS_PER_BLOCK) * Dsz
                        + (size_t)tid * 4;
    const float* hp = h   + base;
    float*       op = out + base;

#pragma unroll 4
    for (int r = 0; r < ROWS_PER_BLOCK; ++r) {
        // Speculative prefetch a few rows ahead (global_prefetch_b8; dropped if unmapped).
        __builtin_prefetch(hp + (size_t)(r + PREFETCH_ROWS) * Dsz, 0, 0);

        v4f x = __builtin_nontemporal_load((const v4f*)(hp + (size_t)r * Dsz));
        x = x * m;
        __builtin_nontemporal_store(x, (v4f*)(op + (size_t)r * Dsz));
    }
}

extern "C" void kernel_launch(void* const* d_in, const int* in_sizes, int n_in,
                              void* d_out, int out_size, void* d_ws, size_t ws_size,
                              hipStream_t stream) {
    const float* h    = (const float*)d_in[0];   // (B, T, D) fp32
    const int*   mask = (const int*)  d_in[1];   // (B, D)   bool -> int32 per harness convention
    float*       out  = (float*)d_out;           // (B, T, D) fp32

    dim3 grid(Bsz * BLOCKS_PER_SAMPLE);          // 2048 blocks
    dim3 block(THREADS);                         // 256 threads (8 waves)
    DropDim_kernel<<<grid, block, 0, stream>>>(h, mask, out);
}